// CTCLayer_17514876633788
// MI455X (gfx1250) — compile-verified
//
#include <hip/hip_runtime.h>
#include <hip/hip_bf16.h>
#include <cstdint>

// Problem constants (from the reference setup_inputs)
#define B_     64
#define T_     1024
#define C_     512
#define L_     128
#define S_     257          // 2L+1
#define BLANK_ (C_ - 1)

#define NEGV  (-1.0e30f)
#define EPSV  (1e-7f)

#define NB     8            // row ring-buffer slots (power of two)
#define DEPTH  7            // async prefetch depth (rows in flight past current)

// ---------------------------------------------------------------------------
// CDNA5 async global->LDS copy (ASYNCcnt-tracked) + wait helpers
// ---------------------------------------------------------------------------
typedef int v4i __attribute__((ext_vector_type(4)));
typedef const __attribute__((address_space(1))) void* gptr_t;
typedef __attribute__((address_space(3))) void*       lptr_t;
typedef __attribute__((address_space(1))) v4i*        gv4_p;
typedef __attribute__((address_space(3))) v4i*        lv4_p;

__device__ __forceinline__ void async_copy_b128(const float* gsrc, float* ldst) {
#if __has_builtin(__builtin_amdgcn_global_load_async_to_lds_b128)
  __builtin_amdgcn_global_load_async_to_lds_b128(
      (gv4_p)(gptr_t)gsrc, (lv4_p)(lptr_t)ldst, /*offset=*/0, /*cpol=*/0);
#else
  unsigned int laddr = (unsigned int)(uintptr_t)(lptr_t)ldst;
  asm volatile("global_load_async_to_lds_b128 %0, %1, off"
               :: "v"(laddr), "v"((unsigned long long)(uintptr_t)gsrc)
               : "memory");
#endif
}

template <int N>
__device__ __forceinline__ void wait_asynccnt() {
#if __has_builtin(__builtin_amdgcn_s_wait_asynccnt)
  __builtin_amdgcn_s_wait_asynccnt(N);
#else
  asm volatile("s_wait_asynccnt %0" :: "n"(N) : "memory");
#endif
}

// logaddexp matching jnp semantics for finite fp32 with NEG sentinels
__device__ __forceinline__ float lae(float a, float b) {
  float m = fmaxf(a, b);
  return m + log1pf(__expf(-fabsf(a - b)));
}

// ---------------------------------------------------------------------------
// Phase 1: lse[b,t] = log( sum_c y_pred[b,t,c] + C*eps )
// (log_softmax(log(y+eps)) == log(y+eps) - lse)   — pure bandwidth kernel.
// One wave32 per row, float4 (b128) loads, shuffle reduction.
// ---------------------------------------------------------------------------
__global__ __launch_bounds__(256) void ctc_lse_kernel(
    const float* __restrict__ yp, float* __restrict__ lse) {
  const int row  = blockIdx.x * 8 + (threadIdx.x >> 5);
  const int lane = threadIdx.x & 31;
  const float4* p = (const float4*)(yp + (size_t)row * C_);
  float s = 0.f;
#pragma unroll
  for (int k = 0; k < 4; ++k) {
    float4 v = p[lane + 32 * k];
    s += (v.x + v.y) + (v.z + v.w);
  }
#pragma unroll
  for (int off = 16; off; off >>= 1) s += __shfl_xor(s, off, 32);
  if (lane == 0) lse[row] = logf(s + (float)C_ * EPSV);
}

// ---------------------------------------------------------------------------
// Phase 2: CTC forward DP, one block (9 wave32s) per batch element.
//  - alpha ping-pong in LDS, one barrier per time step
//  - y_pred class rows (2KB) streamed into an 8-slot LDS ring via async
//    load-to-LDS, depth-7 prefetch hides HBM/L2 latency off the serial chain
//  - lse row preloaded into LDS so no global access on the critical path
// ---------------------------------------------------------------------------
__global__ __launch_bounds__(288) void ctc_dp_kernel(
    const int*   __restrict__ ytrue,   // [B,L]
    const float* __restrict__ yp,      // [B,T,C]
    const float* __restrict__ lse,     // [B,T]
    float*       __restrict__ out) {   // [B]
  __shared__ float rows[NB][C_];       // 16 KB ring of class rows
  __shared__ float alpha[2][S_ + 2];   // two NEG pads at the front
  __shared__ float lseL[T_];           // 4 KB
  __shared__ int   lab[S_];
  __shared__ unsigned char skipf[S_];

  const int b   = blockIdx.x;
  const int tid = threadIdx.x;
  const float* ypb = yp + (size_t)b * T_ * C_;

  // --- extended sequence + skip mask --------------------------------------
  if (tid < S_) {
    int l, sk;
    if (tid & 1) {
      const int i = tid >> 1;
      l  = ytrue[b * L_ + i];
      sk = (i == 0) ? 1 : (l != ytrue[b * L_ + i - 1]);
    } else {
      l = BLANK_; sk = 0;
    }
    lab[tid]   = l;
    skipf[tid] = (unsigned char)sk;
  }
  if (tid < 2) { alpha[0][tid] = NEGV; alpha[1][tid] = NEGV; }
  for (int i = tid; i < T_; i += 288) lseL[i] = lse[b * T_ + i];

  // --- prologue: launch rows 0..DEPTH-1 (128 lanes x 16B == one 2KB row) --
  if (tid < 128) {
#pragma unroll
    for (int r = 0; r < DEPTH; ++r)
      async_copy_b128(ypb + r * C_ + tid * 4, &rows[r][tid * 4]);
    wait_asynccnt<DEPTH - 1>();        // row 0 complete (in-order)
  }
  __syncthreads();                     // row 0 + lab/skip/lse visible

  // --- alpha_0: only states 0,1 reachable ---------------------------------
  if (tid < S_) {
    float a0 = NEGV;
    if (tid < 2) a0 = __logf(rows[0][lab[tid]] + EPSV) - lseL[0];
    alpha[0][2 + tid] = a0;
  }
  if (tid < 128) {
    async_copy_b128(ypb + DEPTH * C_ + tid * 4, &rows[DEPTH][tid * 4]);
    wait_asynccnt<DEPTH - 1>();        // row 1 complete
  }
  __syncthreads();

  // --- DP over t = 1..T-1 --------------------------------------------------
  for (int t = 1; t < T_; ++t) {
    const int slot = t & (NB - 1);
    const int cur  = t & 1;
    const int prv  = cur ^ 1;
    if (tid < S_) {
      const float e  = __logf(rows[slot][lab[tid]] + EPSV) - lseL[t];
      const float a  = alpha[prv][2 + tid];
      const float a1 = alpha[prv][1 + tid];
      const float a2 = skipf[tid] ? alpha[prv][tid] : NEGV;
      alpha[cur][2 + tid] = lae(lae(a, a1), a2) + e;
    }
    if (tid < 128) {
      const int tn = t + DEPTH;        // slot (t-1)&7: row t-1, retired
      if (tn < T_)
        async_copy_b128(ypb + (size_t)tn * C_ + tid * 4,
                        &rows[tn & (NB - 1)][tid * 4]);
      if (t <= T_ - 1 - DEPTH) wait_asynccnt<DEPTH - 1>();  // row t+1 done
      else                     wait_asynccnt<0>();          // tail drain
    }
    __syncthreads();
  }

  // --- loss = -logaddexp(alpha_T[S-1], alpha_T[S-2]) ----------------------
  if (tid == 0) {
    const float x = alpha[(T_ - 1) & 1][2 + S_ - 1];
    const float y = alpha[(T_ - 1) & 1][2 + S_ - 2];
    out[b] = -lae(x, y);
  }
}

// ---------------------------------------------------------------------------
extern "C" void kernel_launch(void* const* d_in, const int* in_sizes, int n_in,
                              void* d_out, int out_size, void* d_ws,
                              size_t ws_size, hipStream_t stream) {
  (void)in_sizes; (void)n_in; (void)out_size; (void)ws_size;
  const int*   y_true = (const int*)d_in[0];    // [B,L] int32
  const float* y_pred = (const float*)d_in[1];  // [B,T,C] fp32
  float* lse = (float*)d_ws;                    // B*T floats = 256 KB scratch
  float* out = (float*)d_out;                   // [B,1] fp32

  ctc_lse_kernel<<<(B_ * T_) / 8, 256, 0, stream>>>(y_pred, lse);
  ctc_dp_kernel<<<B_, 288, 0, stream>>>(y_true, y_pred, lse, out);
}